// AttentionDecoder_257698038454
// MI455X (gfx1250) — compile-verified
//
#include <hip/hip_runtime.h>
#include <hip/hip_bf16.h>
#include <math.h>

// Problem constants (match reference)
#define BB   128
#define TT   50
#define VV   10000
#define WDIM 512
#define AA   512
#define HH   512
#define DD   2048
#define NNIMG 49          // GH*GW
#define XD   (DD + WDIM)  // 2560
#define G3H  (3 * HH)     // 1536

// GEMM pipeline shape
#define UF      2              // K-chunks of 4 per stage
#define STAGE_K (UF * 4)       // 8
#define PAIR_K  (2 * STAGE_K)  // 16; K must be divisible by 16 (true here)

typedef __attribute__((ext_vector_type(2))) float v2f;
typedef __attribute__((ext_vector_type(8))) float v8f;

// ---------------------------------------------------------------------------
// Stable descending argsort of cap_lens (B=128, one block)
// ---------------------------------------------------------------------------
__global__ void sort_lens_kernel(const int* __restrict__ cap_lens,
                                 int* __restrict__ order,
                                 int* __restrict__ slen) {
  int i = threadIdx.x;
  if (i < BB) {
    int li = cap_lens[i];
    int rank = 0;
    for (int j = 0; j < BB; ++j) {
      int lj = cap_lens[j];
      if (lj > li || (lj == li && j < i)) rank++;
    }
    order[rank] = i;
  }
  __syncthreads();
  if (i < BB) slen[i] = cap_lens[order[i]];
}

// ---------------------------------------------------------------------------
// (B,D,GH,GW) -> sorted (B,N,D) + mean over N
// ---------------------------------------------------------------------------
__global__ void permute_mean_kernel(const float* __restrict__ image_code,
                                    const int* __restrict__ order,
                                    float* __restrict__ img_s,
                                    float* __restrict__ meanimg) {
  int b = blockIdx.y;
  int d = blockIdx.x * blockDim.x + threadIdx.x;
  if (d >= DD) return;
  int o = order[b];
  const float* src = image_code + ((size_t)o * DD + d) * NNIMG;
  float s = 0.f;
  for (int n = 0; n < NNIMG; ++n) {
    float v = src[n];
    img_s[((size_t)b * NNIMG + n) * DD + d] = v;
    s += v;
  }
  meanimg[(size_t)b * DD + d] = s * (1.0f / NNIMG);
}

// ---------------------------------------------------------------------------
// one-time weight transpose: WT[n*K + k] = W[k*N + n]  (W is KxN row-major)
// ---------------------------------------------------------------------------
__global__ void transpose_kernel(const float* __restrict__ W,
                                 float* __restrict__ WT, int K, int N) {
  int n = blockIdx.x * blockDim.x + threadIdx.x;
  int k = blockIdx.y;
  if (n < N) WT[(size_t)n * K + k] = W[(size_t)k * N + n];
}

// ---------------------------------------------------------------------------
// WMMA f32 GEMM: C(MxN) = A(MxK) * BT^T + bias, BT row-major NxK.
// Double-buffered stages of UF K-chunks; integer-offset addressing off the
// __restrict__ kernel args (keeps global address space -> global_load_b64).
// A-fragment layout (ISA 7.12.2, 32-bit A 16x4): lanes 0-15 hold K=k0+0,k0+1;
// lanes 16-31 hold K=k0+2,k0+3, row M = lane&15. B mirrored with N = lane&15.
// C layout: VGPR r -> M = r + 8*(lane>=16), N = lane&15.
// ---------------------------------------------------------------------------
struct Stage {
  v2f a[UF];
  v2f b[UF][4];
};

__device__ __forceinline__ void load_stage(Stage& s,
                                           const float* __restrict__ A,
                                           const float* __restrict__ Bm,
                                           size_t& aoff, size_t (&boff)[4]) {
#pragma unroll
  for (int u = 0; u < UF; ++u) {
    float2 av = *reinterpret_cast<const float2*>(&A[aoff + u * 4]);
    s.a[u].x = av.x;
    s.a[u].y = av.y;
#pragma unroll
    for (int j = 0; j < 4; ++j) {
      float2 bv = *reinterpret_cast<const float2*>(&Bm[boff[j] + u * 4]);
      s.b[u][j].x = bv.x;
      s.b[u][j].y = bv.y;
    }
  }
  aoff += STAGE_K;
#pragma unroll
  for (int j = 0; j < 4; ++j) boff[j] += STAGE_K;
}

__device__ __forceinline__ void compute_stage(const Stage& s, v8f acc[4]) {
#pragma unroll
  for (int u = 0; u < UF; ++u) {
#pragma unroll
    for (int j = 0; j < 4; ++j) {
      acc[j] = __builtin_amdgcn_wmma_f32_16x16x4_f32(
          false, s.a[u], false, s.b[u][j], (short)0, acc[j], false, false);
    }
  }
}

// EPI=0: C[m*ldc+n] = acc + bias[n]
// EPI=1: preds epilogue -> P[b*T*V + t*V + n] = (acc+bias)*mask(b,t)
// One wave computes a 16x64 C band (4 tiles); block = 8 waves (16x512 band).
template <int EPI>
__global__ __launch_bounds__(256, 1) void gemm_f32_wmma(
    const float* __restrict__ A, const float* __restrict__ Bm,
    const float* __restrict__ bias, float* __restrict__ C,
    int M, int N, int K, int lda, int ldb, int ldc,
    const int* __restrict__ slen, const int* __restrict__ caplen, int t) {
  const int lane = threadIdx.x & 31;
  const int wid  = threadIdx.x >> 5;
  const int half = lane >> 4;       // 0: K +0/+1, 1: K +2/+3
  const int l16  = lane & 15;
  const int mt   = blockIdx.y;                  // 16-row band
  const int n0   = (blockIdx.x * 8 + wid) * 64; // first of 64 cols

  // branchless column addressing: clamp OOB tiles, never store them
  int ncol[4];
  bool nvalid[4];
#pragma unroll
  for (int j = 0; j < 4; ++j) {
    int n = n0 + j * 16 + l16;
    nvalid[j] = (n0 + j * 16) < N;  // N % 16 == 0 -> tile fully in or out
    ncol[j] = n < N ? n : (N - 1);
  }

  // integer element offsets (half-offset baked in)
  size_t aoff = (size_t)(mt * 16 + l16) * lda + half * 2;
  size_t boff[4];
#pragma unroll
  for (int j = 0; j < 4; ++j) boff[j] = (size_t)ncol[j] * ldb + half * 2;

  v8f acc[4] = {v8f{}, v8f{}, v8f{}, v8f{}};

  Stage s0, s1;
  load_stage(s0, A, Bm, aoff, boff);
  for (int k0 = 0; k0 < K; k0 += PAIR_K) {
    load_stage(s1, A, Bm, aoff, boff);
    compute_stage(s0, acc);
    if (k0 + PAIR_K < K) load_stage(s0, A, Bm, aoff, boff);
    compute_stage(s1, acc);
  }

#pragma unroll
  for (int j = 0; j < 4; ++j) {
    if (!nvalid[j]) continue;
    int n = n0 + j * 16 + l16;
    float bv = bias ? bias[n] : 0.f;
#pragma unroll
    for (int r = 0; r < 8; ++r) {
      int mm = mt * 16 + half * 8 + r;
      float v = acc[j][r] + bv;
      if (EPI == 1) {
        // preds masking: active (t < slen-1) AND unsorted-cap_lens tail mask
        float sc = ((t < slen[mm] - 1) && (t < caplen[mm])) ? 1.f : 0.f;
        C[(size_t)mm * TT * VV + (size_t)t * VV + n] = v * sc;
      } else {
        C[(size_t)mm * ldc + n] = v;
      }
    }
  }
}

// ---------------------------------------------------------------------------
// scores[b,n] = sum_a tanh(q[b,a] + kproj[b,n,a]) * w2[a] + b2   (one wave each)
// ---------------------------------------------------------------------------
__global__ void scores_kernel(const float* __restrict__ q,
                              const float* __restrict__ kproj,
                              const float* __restrict__ w2,
                              const float* __restrict__ b2,
                              float* __restrict__ scores) {
  int wid = threadIdx.x >> 5;
  int lane = threadIdx.x & 31;
  int idx = blockIdx.x * 8 + wid;  // b*NN + n
  if (idx >= BB * NNIMG) return;
  int b = idx / NNIMG;
  const float* qr = q + (size_t)b * AA;
  const float* kr = kproj + (size_t)idx * AA;
  float s = 0.f;
  for (int a = lane; a < AA; a += 32) s += tanhf(qr[a] + kr[a]) * w2[a];
  for (int off = 16; off; off >>= 1) s += __shfl_xor(s, off, 32);
  if (lane == 0) scores[idx] = s + b2[0];
}

// ---------------------------------------------------------------------------
// softmax over N=49 per batch row; also writes masked alpha into Al output
// ---------------------------------------------------------------------------
__global__ void softmax_alpha_kernel(const float* __restrict__ scores,
                                     float* __restrict__ alpha,
                                     float* __restrict__ Al,
                                     const int* __restrict__ slen, int t) {
  int b = blockIdx.x;
  int lane = threadIdx.x;  // 32 threads
  float v0 = (lane < NNIMG) ? scores[b * NNIMG + lane] : -1e30f;
  float v1 = (lane + 32 < NNIMG) ? scores[b * NNIMG + lane + 32] : -1e30f;
  float m = fmaxf(v0, v1);
  for (int off = 16; off; off >>= 1) m = fmaxf(m, __shfl_xor(m, off, 32));
  float e0 = (lane < NNIMG) ? expf(v0 - m) : 0.f;
  float e1 = (lane + 32 < NNIMG) ? expf(v1 - m) : 0.f;
  float s = e0 + e1;
  for (int off = 16; off; off >>= 1) s += __shfl_xor(s, off, 32);
  float inv = 1.f / s;
  float act = (t < slen[b] - 1) ? 1.f : 0.f;
  size_t albase = ((size_t)b * TT + t) * NNIMG;
  if (lane < NNIMG) {
    float a0 = e0 * inv;
    alpha[b * NNIMG + lane] = a0;
    Al[albase + lane] = a0 * act;
  }
  if (lane + 32 < NNIMG) {
    float a1 = e1 * inv;
    alpha[b * NNIMG + lane + 32] = a1;
    Al[albase + lane + 32] = a1 * act;
  }
}

// ---------------------------------------------------------------------------
// ctx[b,d] = sum_n alpha[b,n]*img_s[b,n,d]  -> x[b, 0..D)
// ---------------------------------------------------------------------------
__global__ void ctx_kernel(const float* __restrict__ alpha,
                           const float* __restrict__ img_s,
                           float* __restrict__ x) {
  int b = blockIdx.y;
  int d = blockIdx.x * blockDim.x + threadIdx.x;
  if (d >= DD) return;
  float s = 0.f;
  for (int n = 0; n < NNIMG; ++n)
    s += alpha[b * NNIMG + n] * img_s[((size_t)b * NNIMG + n) * DD + d];
  x[(size_t)b * XD + d] = s;
}

// ---------------------------------------------------------------------------
// embedding gather -> x[b, D..D+WD)
// ---------------------------------------------------------------------------
__global__ void emb_kernel(const int* __restrict__ captions,
                           const int* __restrict__ order,
                           const float* __restrict__ embW,
                           float* __restrict__ x, int t) {
  int b = blockIdx.y;
  int j = blockIdx.x * blockDim.x + threadIdx.x;
  if (j >= WDIM) return;
  int cap = captions[order[b] * TT + t];
  x[(size_t)b * XD + DD + j] = embW[(size_t)cap * WDIM + j];
}

// ---------------------------------------------------------------------------
// GRU cell elementwise + shrinking-batch state update
// ---------------------------------------------------------------------------
__global__ void gru_kernel(const float* __restrict__ gi,
                           const float* __restrict__ gh,
                           float* __restrict__ h,
                           float* __restrict__ hnew,
                           const int* __restrict__ slen, int t) {
  int b = blockIdx.y;
  int k = blockIdx.x * blockDim.x + threadIdx.x;
  if (k >= HH) return;
  size_t base = (size_t)b * G3H;
  float ir = gi[base + k], iz = gi[base + HH + k], inn = gi[base + 2 * HH + k];
  float hr = gh[base + k], hz = gh[base + HH + k], hn = gh[base + 2 * HH + k];
  float r = 1.f / (1.f + expf(-(ir + hr)));
  float z = 1.f / (1.f + expf(-(iz + hz)));
  float n = tanhf(inn + r * hn);
  float hp = h[(size_t)b * HH + k];
  float hv = (1.f - z) * n + z * hp;
  hnew[(size_t)b * HH + k] = hv;
  h[(size_t)b * HH + k] = (t < slen[b] - 1) ? hv : hp;
}

// ---------------------------------------------------------------------------
// zero the padded t = T-1 slices of P and Al
// ---------------------------------------------------------------------------
__global__ void zero_tail_kernel(float* __restrict__ P, float* __restrict__ Al) {
  size_t i = (size_t)blockIdx.x * blockDim.x + threadIdx.x;
  if (i < (size_t)BB * VV) {
    size_t b = i / VV, v = i % VV;
    P[(b * TT + (TT - 1)) * (size_t)VV + v] = 0.f;
  }
  if (i < (size_t)BB * NNIMG) {
    size_t b = i / NNIMG, n = i % NNIMG;
    Al[(b * TT + (TT - 1)) * (size_t)NNIMG + n] = 0.f;
  }
}

// ---------------------------------------------------------------------------
extern "C" void kernel_launch(void* const* d_in, const int* in_sizes, int n_in,
                              void* d_out, int out_size, void* d_ws, size_t ws_size,
                              hipStream_t stream) {
  const float* image_code = (const float*)d_in[0];
  const int*   captions   = (const int*)d_in[1];
  const int*   cap_lens   = (const int*)d_in[2];
  const float* embed_W    = (const float*)d_in[3];
  const float* Wq  = (const float*)d_in[4];
  const float* bq  = (const float*)d_in[5];
  const float* Wk  = (const float*)d_in[6];
  const float* bk  = (const float*)d_in[7];
  const float* w2  = (const float*)d_in[8];
  const float* b2  = (const float*)d_in[9];
  const float* Wis = (const float*)d_in[10];
  const float* bis = (const float*)d_in[11];
  const float* Wih = (const float*)d_in[12];
  const float* Whh = (const float*)d_in[13];
  const float* bih = (const float*)d_in[14];
  const float* bhh = (const float*)d_in[15];
  const float* fcW = (const float*)d_in[16];
  const float* fcb = (const float*)d_in[17];
  (void)in_sizes; (void)n_in; (void)out_size; (void)ws_size;

  float* P  = (float*)d_out;
  float* Al = P + (size_t)BB * TT * VV;

  // workspace carve (256B aligned)
  char* w = (char*)d_ws;
  auto carve = [&](size_t bytes) -> char* {
    char* p = w;
    w += (bytes + 255) & ~(size_t)255;
    return p;
  };
  float* img_s   = (float*)carve(sizeof(float) * (size_t)BB * NNIMG * DD);
  float* kproj   = (float*)carve(sizeof(float) * (size_t)BB * NNIMG * AA);
  float* meanimg = (float*)carve(sizeof(float) * (size_t)BB * DD);
  float* h       = (float*)carve(sizeof(float) * (size_t)BB * HH);
  float* hnew    = (float*)carve(sizeof(float) * (size_t)BB * HH);
  float* q       = (float*)carve(sizeof(float) * (size_t)BB * AA);
  float* scores  = (float*)carve(sizeof(float) * (size_t)BB * NNIMG);
  float* alpha   = (float*)carve(sizeof(float) * (size_t)BB * NNIMG);
  float* x       = (float*)carve(sizeof(float) * (size_t)BB * XD);
  float* gi      = (float*)carve(sizeof(float) * (size_t)BB * G3H);
  float* gh      = (float*)carve(sizeof(float) * (size_t)BB * G3H);
  float* WqT     = (float*)carve(sizeof(float) * (size_t)AA * HH);   // 512x512
  float* WkT     = (float*)carve(sizeof(float) * (size_t)AA * DD);   // 512x2048
  float* WisT    = (float*)carve(sizeof(float) * (size_t)HH * DD);   // 512x2048
  int*   order   = (int*)carve(sizeof(int) * BB);
  int*   slen    = (int*)carve(sizeof(int) * BB);

  // ---- precompute ----
  sort_lens_kernel<<<1, BB, 0, stream>>>(cap_lens, order, slen);
  permute_mean_kernel<<<dim3(DD / 256, BB), 256, 0, stream>>>(image_code, order,
                                                              img_s, meanimg);
  // one-time weight transposes so every GEMM uses the contiguous-B path
  transpose_kernel<<<dim3(AA / 256, HH), 256, 0, stream>>>(Wq, WqT, HH, AA);
  transpose_kernel<<<dim3(AA / 256, DD), 256, 0, stream>>>(Wk, WkT, DD, AA);
  transpose_kernel<<<dim3(HH / 256, DD), 256, 0, stream>>>(Wis, WisT, DD, HH);

  // h0 = meanimg @ Wis + bis   (128 x 512 x 2048)
  gemm_f32_wmma<0><<<dim3(1, BB / 16), 256, 0, stream>>>(
      meanimg, WisT, bis, h, BB, HH, DD, DD, DD, HH, nullptr, nullptr, 0);
  // kproj = img_s @ Wk + bk    (6272 x 512 x 2048)
  gemm_f32_wmma<0><<<dim3(1, (BB * NNIMG) / 16), 256, 0, stream>>>(
      img_s, WkT, bk, kproj, BB * NNIMG, AA, DD, DD, DD, AA, nullptr, nullptr, 0);
  zero_tail_kernel<<<(BB * VV + 255) / 256, 256, 0, stream>>>(P, Al);

  // ---- sequential decode ----
  for (int t = 0; t < TT - 1; ++t) {
    // q = h @ Wq + bq          (128 x 512 x 512)
    gemm_f32_wmma<0><<<dim3(1, BB / 16), 256, 0, stream>>>(
        h, WqT, bq, q, BB, AA, HH, HH, HH, AA, nullptr, nullptr, 0);
    scores_kernel<<<(BB * NNIMG + 7) / 8, 256, 0, stream>>>(q, kproj, w2, b2,
                                                            scores);
    softmax_alpha_kernel<<<BB, 32, 0, stream>>>(scores, alpha, Al, slen, t);
    ctx_kernel<<<dim3(DD / 256, BB), 256, 0, stream>>>(alpha, img_s, x);
    emb_kernel<<<dim3(WDIM / 256, BB), 256, 0, stream>>>(captions, order,
                                                         embed_W, x, t);
    // gi = x @ Wih^T + bih     (128 x 1536 x 2560)
    gemm_f32_wmma<0><<<dim3((G3H + 511) / 512, BB / 16), 256, 0, stream>>>(
        x, Wih, bih, gi, BB, G3H, XD, XD, XD, G3H, nullptr, nullptr, 0);
    // gh = h @ Whh^T + bhh     (128 x 1536 x 512)
    gemm_f32_wmma<0><<<dim3((G3H + 511) / 512, BB / 16), 256, 0, stream>>>(
        h, Whh, bhh, gh, BB, G3H, HH, HH, HH, G3H, nullptr, nullptr, 0);
    gru_kernel<<<dim3(HH / 256, BB), 256, 0, stream>>>(gi, gh, h, hnew, slen, t);
    // preds = hnew @ fcW^T + fcb, fused masks, direct write into P
    gemm_f32_wmma<1><<<dim3((VV + 511) / 512, BB / 16), 256, 0, stream>>>(
        hnew, fcW, fcb, P, BB, VV, HH, HH, HH, 0, slen, cap_lens, t);
  }
}